// PiecewiseForwardNet_13408887898966
// MI455X (gfx1250) — compile-verified
//
#include <hip/hip_runtime.h>

// Problem constants (match reference)
#define N_ROWS 16384
#define OBS    512
#define ENC    512
#define ACT    64
#define NEXP   8

typedef float v2f __attribute__((ext_vector_type(2)));
typedef float v8f __attribute__((ext_vector_type(8)));
typedef unsigned int v4u __attribute__((ext_vector_type(4)));
typedef int v4i __attribute__((ext_vector_type(4)));
typedef int v8i __attribute__((ext_vector_type(8)));

// LDS row strides (padded so gathered rows land 4 banks apart -> conflict-free
// ds_load_b64 A-fragment reads). Matches TDM pad: 2 dwords per 256 / per 64.
#define X0_LDS_STRIDE 516   // 512 + 2 (pad @256) + 2 (pad @512)
#define U_LDS_STRIDE  66    // 64 + 2

// D = A(16x4, f32) * B(4x16, f32) + C ; CDNA5 V_WMMA_F32_16X16X4_F32
__device__ __forceinline__ v8f wmma_f32_k4(v2f a, v2f b, v8f c) {
  return __builtin_amdgcn_wmma_f32_16x16x4_f32(
      /*neg_a=*/false, a, /*neg_b=*/false, b,
      /*c_mod=*/(short)0, c, /*reuse_a=*/false, /*reuse_b=*/false);
}

// TDM gather: DMA 16 rows (16-bit indices) of a row-major [tensor_rows x width]
// f32 tensor into LDS, with optional padding inserted every (2<<pad_int) dwords.
// D# layout per CDNA5 ISA §8.3-8.7 (gather mode).
__device__ __forceinline__ void tdm_gather_rows_f32(
    unsigned lds_addr, const float* base, unsigned width, unsigned stride,
    unsigned tensor_rows, unsigned pad_en, unsigned pad_int_code,
    unsigned pad_amt_code, const int* rr) {
  unsigned long long ga = (unsigned long long)(size_t)base;
  v4u g0;
  g0.x = 0x80000001u;                                   // count=1, gather_mode=1, 16b idx
  g0.y = lds_addr;                                      // LDS byte address
  g0.z = (unsigned)ga;                                  // global_addr[31:0]
  g0.w = (unsigned)((ga >> 32) & 0x1FFFFFFu) | (2u << 30);  // addr[56:32] | type=2
  v8i g1;
  g1[0] = (int)((2u << 16) | (pad_en << 20) | (pad_int_code << 22) |
                (pad_amt_code << 25));                  // data_size=4B, pad cfg
  g1[1] = (int)((width & 0xFFFFu) << 16);               // tensor_dim0[15:0]
  g1[2] = (int)(((width >> 16) & 0xFFFFu) | ((tensor_rows & 0xFFFFu) << 16));
  g1[3] = (int)(((tensor_rows >> 16) & 0xFFFFu) | ((width & 0xFFFFu) << 16)); // tile_dim0
  g1[4] = 16;                                           // tile_dim1 = #indices
  g1[5] = (int)stride;                                  // tensor_dim0_stride[31:0]
  g1[6] = 0;
  g1[7] = 0;
  v4i g2, g3;
#pragma unroll
  for (int i = 0; i < 4; ++i) {
    g2[i] = (rr[2 * i] & 0xFFFF) | (rr[2 * i + 1] << 16);
    g3[i] = (rr[8 + 2 * i] & 0xFFFF) | (rr[8 + 2 * i + 1] << 16);
  }
#if __clang_major__ >= 23
  __builtin_amdgcn_tensor_load_to_lds(g0, g1, g2, g3, g1, 0);
#else
  __builtin_amdgcn_tensor_load_to_lds(g0, g1, g2, g3, 0);
#endif
}

// ---------------------------------------------------------------------------
// Kernel 0: zero routing counters and the output accumulator.
// ---------------------------------------------------------------------------
__global__ void init_kernel(int* __restrict__ counts, float* __restrict__ out) {
  if (threadIdx.x < NEXP) counts[threadIdx.x] = 0;
  if (threadIdx.x == 0) out[0] = 0.0f;
}

// ---------------------------------------------------------------------------
// Kernel 1: encoder GEMM  Y = X @ W^T  for X in {X0, X1} (blockIdx.z selects).
// 8 waves per block; each wave owns a 16x64 strip (4 accumulators): one A
// fragment is reused across 4 B fragments per k-step -> 16 wmma / 20 loads
// per unrolled iteration.
// ---------------------------------------------------------------------------
__global__ __launch_bounds__(256)
void enc_gemm_kernel(const float* __restrict__ X0, const float* __restrict__ X1,
                     const float* __restrict__ W, float* __restrict__ X0e,
                     float* __restrict__ X1e) {
  const int wave  = threadIdx.x >> 5;
  const int lane  = threadIdx.x & 31;
  const int laneM = lane & 15;
  const int kk0   = (lane >> 4) << 1;              // 0 | 2
  const int e0    = blockIdx.x * 64;               // 64-wide output strip
  const int m0    = (blockIdx.y * 8 + wave) * 16;  // row tile

  const float* X = blockIdx.z ? X1 : X0;
  float*       Y = blockIdx.z ? X1e : X0e;

  const float* arow = X + (size_t)(m0 + laneM) * OBS + kk0;
  const float* brow = W + (size_t)(e0 + laneM) * OBS + kk0;  // + j*16*OBS per tile

  v8f acc[4] = {};
#pragma unroll 4
  for (int k = 0; k < OBS; k += 4) {
    v2f a = *(const v2f*)(arow + k);
#pragma unroll
    for (int j = 0; j < 4; ++j) {
      v2f b = *(const v2f*)(brow + (size_t)j * 16 * OBS + k);
      acc[j] = wmma_f32_k4(a, b, acc[j]);
    }
  }

  const int rbase = m0 + ((lane >> 4) << 3);
#pragma unroll
  for (int j = 0; j < 4; ++j) {
    const int col = e0 + j * 16 + laneM;
#pragma unroll
    for (int v = 0; v < 8; ++v)
      Y[(size_t)(rbase + v) * ENC + col] = acc[j][v];
  }
}

// ---------------------------------------------------------------------------
// Kernel 2: router. One wave per row; shfl-xor reduce; first-index argmax;
// atomic compaction into per-expert row lists.
// ---------------------------------------------------------------------------
__global__ __launch_bounds__(256)
void route_kernel(const float* __restrict__ X0e, const float* __restrict__ Cw,
                  const float* __restrict__ Cb, int* __restrict__ counts,
                  int* __restrict__ rowlist) {
  const int wave = threadIdx.x >> 5;
  const int lane = threadIdx.x & 31;
  const int n    = blockIdx.x * 8 + wave;

  float acc[NEXP];
#pragma unroll
  for (int j = 0; j < NEXP; ++j) acc[j] = 0.0f;

  const float* xr = X0e + (size_t)n * ENC;
  for (int h = lane; h < ENC; h += 32) {
    const float x = xr[h];
#pragma unroll
    for (int j = 0; j < NEXP; ++j) acc[j] += x * Cw[j * ENC + h];
  }
#pragma unroll
  for (int j = 0; j < NEXP; ++j) {
#pragma unroll
    for (int off = 16; off > 0; off >>= 1)
      acc[j] += __shfl_xor(acc[j], off, 32);
  }
  if (lane == 0) {
    int best = 0;
    float bl = acc[0] + Cb[0];
#pragma unroll
    for (int j = 1; j < NEXP; ++j) {
      const float l = acc[j] + Cb[j];
      if (l > bl) { bl = l; best = j; }
    }
    const int pos = atomicAdd(&counts[best], 1);
    rowlist[best * N_ROWS + pos] = n;
  }
}

// ---------------------------------------------------------------------------
// Kernel 3: grouped expert GEMM + fused squared-error loss.
// Block = (expert e, tile of 16 routed rows). Wave 0 TDM-gathers the 16 X0e
// rows (32KB) and 16 U rows (4KB) into padded LDS; all 8 waves then compute a
// 16x64 strip each (4 accumulators) with ds_load_b64 A-fragments.
// ---------------------------------------------------------------------------
__global__ __launch_bounds__(256)
void expert_loss_kernel(const float* __restrict__ X0e, const float* __restrict__ X1e,
                        const float* __restrict__ U, const float* __restrict__ A_all,
                        const float* __restrict__ B_rest,
                        const int* __restrict__ counts, const int* __restrict__ rowlist,
                        float* __restrict__ out) {
  __shared__ float ldsX0[16 * X0_LDS_STRIDE];
  __shared__ float ldsU[16 * U_LDS_STRIDE];
  __shared__ int   rows_s[16];
  __shared__ float valid_s[16];
  __shared__ float red[256];

  const int e    = blockIdx.y;
  const int cnt  = counts[e];
  const int base = blockIdx.x * 16;
  if (base >= cnt) return;  // uniform per block

  const int wave  = threadIdx.x >> 5;
  const int lane  = threadIdx.x & 31;
  const int laneM = lane & 15;
  const int kk0   = (lane >> 4) << 1;
  const int hi8   = (lane >> 4) << 3;

  if (wave == 0) {
    int rr[16];
#pragma unroll
    for (int i = 0; i < 16; ++i) {
      const int idx = base + i;                       // uniform -> s_load
      rr[i] = rowlist[e * N_ROWS + (idx < cnt ? idx : base)];
    }
    if (lane < 16) {
      rows_s[lane]  = rr[lane];
      valid_s[lane] = (base + lane < cnt) ? 1.0f : 0.0f;
    }
    // gather 16 rows of X0e: pad 2 dwords per 256 dwords (codes 7,1)
    tdm_gather_rows_f32((unsigned)(size_t)ldsX0, X0e, ENC, ENC, N_ROWS,
                        1u, 7u, 1u, rr);
    // gather 16 rows of U: pad 2 dwords per 64 dwords (codes 5,1)
    tdm_gather_rows_f32((unsigned)(size_t)ldsU, U, ACT, ACT, N_ROWS,
                        1u, 5u, 1u, rr);
    __builtin_amdgcn_s_wait_tensorcnt(0);
  }
  __syncthreads();

  const int c0  = wave * 64;       // this wave's 64-col strip
  const int col = c0 + laneM;      // +j*16 per sub-tile

  v8f acc[4] = {};

  // ---- U @ B_e part ----
  if (e == 0) {
    // B0 = eye(ENC)[:ACT]  =>  pred_B[m, c] = (c < ACT) ? U[m, c] : 0
#pragma unroll
    for (int j = 0; j < 4; ++j) {
      const int cc = col + j * 16;
      if (cc < ACT) {
#pragma unroll
        for (int v = 0; v < 8; ++v)
          acc[j][v] = ldsU[(v + hi8) * U_LDS_STRIDE + cc];
      }
    }
  } else {
    const float* Be = B_rest + (size_t)(e - 1) * ACT * ENC;
#pragma unroll 4
    for (int k = 0; k < ACT; k += 4) {
      v2f a = *(const v2f*)&ldsU[laneM * U_LDS_STRIDE + k + kk0];
#pragma unroll
      for (int j = 0; j < 4; ++j) {
        v2f b;
        b.x = Be[(size_t)(k + kk0) * ENC + col + j * 16];
        b.y = Be[(size_t)(k + kk0 + 1) * ENC + col + j * 16];
        acc[j] = wmma_f32_k4(a, b, acc[j]);
      }
    }
  }

  // ---- X0e @ A_e^T part (main) ----
  const float* Ae = A_all + (size_t)e * ENC * ENC + kk0;
#pragma unroll 4
  for (int k = 0; k < ENC; k += 4) {
    const int kp = k + ((k >= 256) ? 2 : 0);  // skip LDS pad at dword 256
    v2f a = *(const v2f*)&ldsX0[laneM * X0_LDS_STRIDE + kp + kk0];
#pragma unroll
    for (int j = 0; j < 4; ++j) {
      v2f b = *(const v2f*)(Ae + (size_t)(col + j * 16) * ENC + k);
      acc[j] = wmma_f32_k4(a, b, acc[j]);
    }
  }

  // ---- fused squared error vs X1e, masked for tail slots ----
  float sq = 0.0f;
#pragma unroll
  for (int j = 0; j < 4; ++j) {
    const int cc = col + j * 16;
#pragma unroll
    for (int v = 0; v < 8; ++v) {
      const int rs = v + hi8;
      const float d = X1e[(size_t)rows_s[rs] * ENC + cc] - acc[j][v];
      sq += d * d * valid_s[rs];
    }
  }

  red[threadIdx.x] = sq;
  __syncthreads();
  for (int s = 128; s > 0; s >>= 1) {
    if (threadIdx.x < s) red[threadIdx.x] += red[threadIdx.x + s];
    __syncthreads();
  }
  if (threadIdx.x == 0)
    atomicAdd(out, red[0] * (1.0f / ((float)ENC * (float)N_ROWS)));  // ALPHA=1
}

// ---------------------------------------------------------------------------
// Launch: init -> encoder GEMM (X0 & X1) -> router -> grouped expert loss.
// Workspace: [X0e: N*ENC f32][X1e: N*ENC f32][counts: K i32 (pad 256B)]
//            [rowlist: K*N i32]   (~67.7 MB total)
// ---------------------------------------------------------------------------
extern "C" void kernel_launch(void* const* d_in, const int* in_sizes, int n_in,
                              void* d_out, int out_size, void* d_ws, size_t ws_size,
                              hipStream_t stream) {
  (void)in_sizes; (void)n_in; (void)out_size; (void)ws_size;

  const float* X1     = (const float*)d_in[0];
  const float* X0     = (const float*)d_in[1];
  const float* U      = (const float*)d_in[2];
  const float* W      = (const float*)d_in[3];
  const float* A_all  = (const float*)d_in[4];
  const float* B_rest = (const float*)d_in[5];
  const float* Cw     = (const float*)d_in[6];
  const float* Cb     = (const float*)d_in[7];
  float* out = (float*)d_out;

  char* ws = (char*)d_ws;
  float* X0e    = (float*)(ws);
  float* X1e    = (float*)(ws + (size_t)N_ROWS * ENC * 4);
  int*   counts = (int*)  (ws + 2ull * N_ROWS * ENC * 4);
  int*   rowlst = (int*)  (ws + 2ull * N_ROWS * ENC * 4 + 256);

  init_kernel<<<dim3(1), dim3(64), 0, stream>>>(counts, out);

  enc_gemm_kernel<<<dim3(ENC / 64, N_ROWS / 128, 2), dim3(256), 0, stream>>>(
      X0, X1, W, X0e, X1e);

  route_kernel<<<dim3(N_ROWS / 8), dim3(256), 0, stream>>>(
      X0e, Cw, Cb, counts, rowlst);

  expert_loss_kernel<<<dim3(N_ROWS / 16, NEXP), dim3(256), 0, stream>>>(
      X0e, X1e, U, A_all, B_rest, counts, rowlst, out);
}